// FP_13511967113529
// MI455X (gfx1250) — compile-verified
//
#include <hip/hip_runtime.h>
#include <math.h>
#include <stdint.h>

#define VIEW_NUM 360
#define BATCH    2
#define IMG      512
#define CHUNK_R  64                       // rows per LDS chunk
#define NCHUNK   (IMG / CHUNK_R)          // 8 chunks
#define BLOCK    512                      // one thread per sinogram column
#define LDS_BYTES (2 * CHUNK_R * IMG * 4) // double-buffered: 256 KB

// ---------------------------------------------------------------------------
// CDNA5 async global->LDS copy (ASYNCcnt-tracked, bypasses VGPRs).
// VDST lane value = LDS byte offset; VADDR = 64-bit global address; saddr off.
// ---------------------------------------------------------------------------
__device__ __forceinline__ void async_g2l_b128(uint32_t lds_byte_off, const void* gaddr) {
    asm volatile("global_load_async_to_lds_b128 %0, %1, off"
                 :: "v"(lds_byte_off), "v"(gaddr)
                 : "memory");
}

__device__ __forceinline__ void wait_asynccnt0() {
    asm volatile("s_wait_asynccnt 0x0" ::: "memory");
}

extern __shared__ float lds[];  // 2 * CHUNK_R * IMG floats, dynamic

__global__ void __launch_bounds__(BLOCK, 1)
fp_radon_kernel(const float* __restrict__ x, float* __restrict__ out) {
    const int tid  = threadIdx.x;          // output column u
    const int b    = blockIdx.x / VIEW_NUM;
    const int view = blockIdx.x % VIEW_NUM;

    const float* __restrict__ img = x + (size_t)b * IMG * IMG;

    // angles = -pi*(i+1)/360 - pi ; fold /360 into a constant multiply and use
    // the hardware trans path (|ang| <= 2*pi, a few-ulp error -> DDA drift
    // < 1e-4 px over 512 steps, far below bilinear resolution).
    const float ang = fmaf((float)(view + 1), -(float)M_PI / (float)VIEW_NUM, -(float)M_PI);
    float sv, cv;
    __sincosf(ang, &sv, &cv);

    // Ray start at t=0; exact unit-step DDA thereafter:
    //   ix(t) = ix0 - sv*t ;  iy(t) = iy0 + cv*t
    const float Xu  = (2.0f * (float)tid + 1.0f) / (float)IMG - 1.0f;
    const float Y0  = 1.0f / (float)IMG - 1.0f;
    const float ix0 = ((cv * Xu - sv * Y0 + 1.0f) * (float)IMG - 1.0f) * 0.5f;
    const float iy0 = ((sv * Xu + cv * Y0 + 1.0f) * (float)IMG - 1.0f) * 0.5f;

    const bool  steep = fabsf(cv) > 1e-6f;
    // Fast v_rcp is safe here: the t-interval it feeds is widened by +-1 and
    // guarded by exact per-step ownership predicates.
    const float inv_c = __builtin_amdgcn_rcpf(cv);

    const uint32_t lds_base = (uint32_t)(uintptr_t)(void*)lds; // LDS byte offset

    // --- issue async loads for chunk 0 ---
    {
        const char* gsrc = (const char*)img;            // rows [0,64)
        #pragma unroll
        for (int k = 0; k < 16; ++k) {
            int f4 = tid + k * BLOCK;                   // float4 index in chunk
            async_g2l_b128(lds_base + (uint32_t)f4 * 16u, gsrc + (size_t)f4 * 16);
        }
    }
    wait_asynccnt0();
    __syncthreads();

    float acc = 0.0f;

    for (int cidx = 0; cidx < NCHUNK; ++cidx) {
        // --- prefetch next chunk into the other buffer (overlaps compute) ---
        if (cidx + 1 < NCHUNK) {
            const char* gsrc = (const char*)(img + (size_t)(cidx + 1) * CHUNK_R * IMG);
            uint32_t dst = lds_base + (uint32_t)(((cidx + 1) & 1) * CHUNK_R * IMG * 4);
            #pragma unroll
            for (int k = 0; k < 16; ++k) {
                int f4 = tid + k * BLOCK;
                async_g2l_b128(dst + (uint32_t)f4 * 16u, gsrc + (size_t)f4 * 16);
            }
        }

        // --- process chunk cidx from current buffer ---
        const float* __restrict__ buf = lds + (cidx & 1) * CHUNK_R * IMG;
        const int r0 = cidx * CHUNK_R;

        // t-interval where floor(iy) in [r0-1, r0+R-1]  <=>  iy in [r0-1, r0+R)
        const float lo = (float)(r0 - 1);
        const float hi = (float)(r0 + CHUNK_R);
        int tlo, thi;
        if (steep) {
            float ta = (lo - iy0) * inv_c;
            float tb = (hi - iy0) * inv_c;
            float tmin = fminf(ta, tb);
            float tmax = fmaxf(ta, tb);
            tlo = max(0,   (int)floorf(tmin) - 1);
            thi = min(IMG, (int)ceilf(tmax) + 1);
        } else {  // near-horizontal ray: iy ~ constant
            bool in = (iy0 >= lo) && (iy0 < hi);
            tlo = 0;
            thi = in ? IMG : 0;
        }

        // Branchless hot loop: 4 ds_load_b32 + FMA/select VALU, no divergence
        // in the body. Row-ownership decomposition: this chunk only adds rows
        // it owns ([r0, r0+R)); out-of-image rows are never owned => zero pad.
        for (int t = tlo; t < thi; ++t) {
            float tf  = (float)t;
            float iy  = fmaf(cv, tf, iy0);
            float ix  = fmaf(-sv, tf, ix0);

            float y0f = floorf(iy);
            float x0f = floorf(ix);
            int   y0  = (int)y0f;
            int   x0  = (int)x0f;
            float wy1 = iy - y0f, wy0 = 1.0f - wy1;
            float wx1 = ix - x0f, wx0 = 1.0f - wx1;

            // x taps shared by both rows (zero-pad in x via weight zeroing)
            int   x1  = x0 + 1;
            bool  vx0 = (unsigned)x0 < (unsigned)IMG;
            bool  vx1 = (unsigned)x1 < (unsigned)IMG;
            int   xi0 = vx0 ? x0 : 0;
            int   xi1 = vx1 ? x1 : 0;
            float wa  = vx0 ? wx0 : 0.0f;
            float wb  = vx1 ? wx1 : 0.0f;

            // row ownership (clamp offset, zero weight)
            int   ry0 = y0 - r0;
            int   ry1 = ry0 + 1;
            bool  vy0 = (unsigned)ry0 < (unsigned)CHUNK_R;
            bool  vy1 = (unsigned)ry1 < (unsigned)CHUNK_R;
            int   o0  = (vy0 ? ry0 : 0) * IMG;
            int   o1  = (vy1 ? ry1 : 0) * IMG;
            float w0  = vy0 ? wy0 : 0.0f;
            float w1  = vy1 ? wy1 : 0.0f;

            float ra = fmaf(wa, buf[o0 + xi0], wb * buf[o0 + xi1]);
            float rb = fmaf(wa, buf[o1 + xi0], wb * buf[o1 + xi1]);
            acc = fmaf(w0, ra, acc);
            acc = fmaf(w1, rb, acc);
        }

        // Drain this wave's prefetch of chunk c+1, then group-sync so all
        // waves agree the next buffer is valid and the old one is reusable.
        wait_asynccnt0();
        __syncthreads();
    }

    // sino(b, 0, u, view) = 0.5 * sum
    out[((size_t)(b * IMG + tid)) * VIEW_NUM + view] = 0.5f * acc;
}

extern "C" void kernel_launch(void* const* d_in, const int* in_sizes, int n_in,
                              void* d_out, int out_size, void* d_ws, size_t ws_size,
                              hipStream_t stream) {
    (void)in_sizes; (void)n_in; (void)out_size; (void)d_ws; (void)ws_size;
    const float* x   = (const float*)d_in[0];
    float*       out = (float*)d_out;

    // 256 KB dynamic LDS per workgroup (gfx1250 WGP has 320 KB).
    hipFuncSetAttribute((const void*)fp_radon_kernel,
                        hipFuncAttributeMaxDynamicSharedMemorySize, LDS_BYTES);

    dim3 grid(BATCH * VIEW_NUM);
    dim3 block(BLOCK);
    fp_radon_kernel<<<grid, block, LDS_BYTES, stream>>>(x, out);
}